// MakeCutouts_7000796692501
// MI455X (gfx1250) — compile-verified
//
#include <hip/hip_runtime.h>
#include <stdint.h>

// MakeCutouts: 32 crops of (8,3,512,512) fp32 -> adaptive_avg_pool2d -> (256,3,224,224)
// Bandwidth-bound (~180 MB HBM traffic => ~8us floor @ 23.3 TB/s). Input (25 MB)
// is L2-resident; per-block crop rows are staged to LDS with CDNA5 async
// global->LDS loads (ASYNCcnt) and outputs are streamed with non-temporal stores.

#define CUTN   32
#define BATCH  8
#define CH     3
#define HW     512
#define OUTSZ  224
#define MAXROWS 4            // max adaptive-pool bin height for sz<=512
#define PITCH  512           // LDS row pitch in floats

#define GLOBAL_AS __attribute__((address_space(1)))
#define LDS_AS    __attribute__((address_space(3)))

#if defined(__HIP_DEVICE_COMPILE__) && __has_builtin(__builtin_amdgcn_global_load_async_to_lds_b32)
#define USE_ASYNC_LDS 1
#else
#define USE_ASYNC_LDS 0
#endif

__global__ __launch_bounds__(OUTSZ) void make_cutouts_pool_kernel(
    const float* __restrict__ x,      // [8,3,512,512]
    const int*   __restrict__ sizes,  // [32]
    const int*   __restrict__ offy,   // [32]
    const int*   __restrict__ offx,   // [32]
    float*       __restrict__ out)    // [256,3,224,224]
{
    __shared__ float tile[MAXROWS * PITCH];   // 8 KB

    const int oy    = blockIdx.x;             // 0..223 output row
    const int plane = blockIdx.y;             // 0..767 = (i*B + b)*C + c
    const int c     = plane % CH;
    const int b     = (plane / CH) % BATCH;
    const int i     = plane / (CH * BATCH);

    // Uniform (blockIdx-only) -> scalar loads
    const int sz = sizes[i];
    const int y0 = offy[i];
    const int x0 = offx[i];

    // Row bin for this output row: [sy, ey) in crop coords (uniform over block)
    const int sy    = (oy * sz) / OUTSZ;
    const int ey    = ((oy + 1) * sz + (OUTSZ - 1)) / OUTSZ;
    const int nrows = ey - sy;                // 1..4

    const int tid = threadIdx.x;              // 0..223

    // ---- Stage nrows x sz crop elements into LDS ------------------------------
    const float* src = x + (((size_t)(b * CH + c) * HW) + (size_t)(y0 + sy)) * HW + x0;
    for (int r = 0; r < nrows; ++r) {
        const float* rowp = src + (size_t)r * HW;
        for (int xx = tid; xx < sz; xx += OUTSZ) {
#if USE_ASYNC_LDS
            // CDNA5 async copy: global -> LDS, per-lane, tracked by ASYNCcnt
            __builtin_amdgcn_global_load_async_to_lds_b32(
                (GLOBAL_AS int*)(rowp + xx),
                (LDS_AS int*)&tile[r * PITCH + xx],
                0, 0);
#else
            tile[r * PITCH + xx] = rowp[xx];
#endif
        }
    }
#if USE_ASYNC_LDS
#if __has_builtin(__builtin_amdgcn_s_wait_asynccnt)
    __builtin_amdgcn_s_wait_asynccnt(0);
#else
    asm volatile("s_wait_asynccnt 0" ::: "memory");
#endif
#endif
    __syncthreads();

    // ---- Per-thread column bin ------------------------------------------------
    const int ox = tid;
    const int sx = (ox * sz) / OUTSZ;
    const int ex = ((ox + 1) * sz + (OUTSZ - 1)) / OUTSZ;   // 1..4 cols

    float sum = 0.0f;
    for (int r = 0; r < nrows; ++r) {
        const float* lrow = &tile[r * PITCH];
        for (int xx = sx; xx < ex; ++xx)
            sum += lrow[xx];
    }
    const float cnt = (float)(nrows * (ex - sx));
    const size_t o  = ((size_t)plane * OUTSZ + (size_t)oy) * OUTSZ + (size_t)ox;

    // Streaming output: written once, never re-read -> non-temporal hint
    __builtin_nontemporal_store(sum / cnt, &out[o]);
}

extern "C" void kernel_launch(void* const* d_in, const int* in_sizes, int n_in,
                              void* d_out, int out_size, void* d_ws, size_t ws_size,
                              hipStream_t stream) {
    (void)in_sizes; (void)n_in; (void)out_size; (void)d_ws; (void)ws_size;
    const float* x     = (const float*)d_in[0];
    const int*   sizes = (const int*)d_in[1];
    const int*   oy    = (const int*)d_in[2];
    const int*   ox    = (const int*)d_in[3];
    // d_in[4] = cut_size scalar (==224, compile-time constant here)
    float* out = (float*)d_out;

    dim3 grid(OUTSZ, CUTN * BATCH * CH);   // (224, 768)
    dim3 block(OUTSZ);                     // 224 threads = 7 wave32
    make_cutouts_pool_kernel<<<grid, block, 0, stream>>>(x, sizes, oy, ox, out);
}